// VanillaEdgeAttnModel_45621142618359
// MI455X (gfx1250) — compile-verified
//
#include <hip/hip_runtime.h>
#include <math.h>

// Problem constants (match reference)
#define E_EDGES 32768
#define K_NBR   32
#define D_DIM   128
#define H_HEADS 4
#define DH_DIM  32
#define C_CLS   10

#define ROWS_PER_BLK 64   // 4 M-tiles of 16
#define MT_TILES     4

typedef __attribute__((ext_vector_type(2))) float v2f;
typedef __attribute__((ext_vector_type(8))) float v8f;
typedef __attribute__((ext_vector_type(4))) unsigned int u32x4;
typedef __attribute__((ext_vector_type(8))) unsigned int u32x8;

// ---------------------------------------------------------------------------
// TDM: stage a contiguous [rows x 128] f32 tile (row stride 128) from global
// into LDS via the Tensor Data Mover.  Descriptor per CDNA5 ISA ch.8 (D#).
// Issued once (TDM ignores EXEC; caller gates to a single wave).
// ---------------------------------------------------------------------------
__device__ __forceinline__ void tdm_load_tile_f32(const float* gsrc,
                                                  float* lds_dst,
                                                  unsigned rows) {
    unsigned long long ga = (unsigned long long)(uintptr_t)gsrc;
    unsigned lds_addr = (unsigned)(uintptr_t)lds_dst;   // low 32 bits = LDS offset

    u32x4 g0;
    g0[0] = 0x1u;                                   // count=1, user mode
    g0[1] = lds_addr;                               // lds_addr[31:0]
    g0[2] = (unsigned)(ga & 0xffffffffu);           // global_addr[31:0]
    g0[3] = (unsigned)((ga >> 32) & 0x01ffffffu)    // global_addr[56:32]
          | (2u << 30);                             // type = 2 ("image")

    u32x8 g1;
    g1[0] = 2u << 16;                               // data_size = 2 (4 bytes)
    g1[1] = (unsigned)D_DIM << 16;                  // tensor_dim0[15:0] (=128)
    g1[2] = rows << 16;                             // tensor_dim0_hi=0 | tensor_dim1=rows
    g1[3] = (unsigned)D_DIM << 16;                  // tensor_dim1_hi=0 | tile_dim0=128
    g1[4] = rows;                                   // tile_dim1=rows | tile_dim2=0
    g1[5] = (unsigned)D_DIM;                        // tensor_dim0_stride lo (=128)
    g1[6] = 0u;                                     // stride0 hi | tensor_dim1_stride lo
    g1[7] = 0u;                                     // tensor_dim1_stride hi
    asm volatile("tensor_load_to_lds %0, %1" :: "s"(g0), "s"(g1) : "memory");
}

// ---------------------------------------------------------------------------
// WMMA: D(16x16,f32) += A(16x4,f32) x B(4x16,f32)   (CDNA5 ISA 7.12.2 layouts)
//   A: VGPR i, lane-group g (lane>>4): (M = lane&15, K = k0 + i + 2*g)
//   B: VGPR i, lane-group g:           (K = k0 + i + 2*g, N = lane&15)
//   C/D: VGPR i, lane-group g:         (M = i + 8*g, N = lane&15)
// ---------------------------------------------------------------------------
__device__ __forceinline__ v8f wmma_f32_step(v2f a, v2f b, v8f c) {
    return __builtin_amdgcn_wmma_f32_16x16x4_f32(
        /*neg_a=*/false, a, /*neg_b=*/false, b,
        /*c_mod=*/(short)0, c, /*reuse_a=*/false, /*reuse_b=*/false);
}

// acc[mt] += sA(mt-th 16x128 tile, LDS) @ B(128x128 global, cols n0..n0+15)
// B fragments are loaded once per k-step and reused across all 4 M-tiles.
__device__ __forceinline__ void tile_gemm_4m(const float* __restrict__ sA,
                                             const float* __restrict__ B,
                                             int n0, int lane, v8f acc[MT_TILES]) {
    const int mrow = lane & 15;
    const int g    = lane >> 4;
    const int ncol = n0 + (lane & 15);
    #pragma unroll 8
    for (int k0 = 0; k0 < D_DIM; k0 += 4) {
        v2f b;
        b[0] = B[(size_t)(k0 + 2 * g)     * D_DIM + ncol];
        b[1] = B[(size_t)(k0 + 2 * g + 1) * D_DIM + ncol];
        #pragma unroll
        for (int mt = 0; mt < MT_TILES; ++mt) {
            v2f a;
            a[0] = sA[(mt * 16 + mrow) * D_DIM + k0 + 2 * g];
            a[1] = sA[(mt * 16 + mrow) * D_DIM + k0 + 2 * g + 1];
            acc[mt] = wmma_f32_step(a, b, acc[mt]);
        }
    }
}

// ---------------------------------------------------------------------------
// Kernel 1: q / k_all / v_all projections.
// grid.x = E/64, block = 256 (8 waves).  X tile staged by TDM; each wave owns
// one 16-wide N tile and 4 M-tiles; loops over the three weight matrices.
// ---------------------------------------------------------------------------
__global__ void __launch_bounds__(256)
qkv_proj_kernel(const float* __restrict__ X,
                const float* __restrict__ Wq,
                const float* __restrict__ Wk,
                const float* __restrict__ Wv,
                float* __restrict__ q,
                float* __restrict__ k,
                float* __restrict__ v) {
    __shared__ float sX[ROWS_PER_BLK * D_DIM];
    const int row0 = blockIdx.x * ROWS_PER_BLK;

    if (threadIdx.x < 32) {   // single wave issues the DMA
        tdm_load_tile_f32(X + (size_t)row0 * D_DIM, sX, ROWS_PER_BLK);
        __builtin_amdgcn_s_wait_tensorcnt(0);
    }
    __syncthreads();

    const int wave = threadIdx.x >> 5;
    const int lane = threadIdx.x & 31;
    const int n0   = wave * 16;
    const int g    = lane >> 4;
    const int ncol = n0 + (lane & 15);

    const float* Ws[3] = {Wq, Wk, Wv};
    float*       Os[3] = {q, k, v};
    #pragma unroll
    for (int m = 0; m < 3; ++m) {
        v8f acc[MT_TILES] = {};
        tile_gemm_4m(sX, Ws[m], n0, lane, acc);
        float* O = Os[m];
        #pragma unroll
        for (int mt = 0; mt < MT_TILES; ++mt)
            #pragma unroll
            for (int i = 0; i < 8; ++i)
                O[(size_t)(row0 + mt * 16 + i + 8 * g) * D_DIM + ncol] = acc[mt][i];
    }
}

// ---------------------------------------------------------------------------
// Kernel 2: per-edge attention over 32 gathered neighbors.
// grid.x = E, block = 128 (4 waves; wave == head).  Gathers hit L2
// (k_all / v_all are 16.8 MB each << 192 MB L2).
// ---------------------------------------------------------------------------
__global__ void __launch_bounds__(128)
edge_attn_kernel(const float* __restrict__ q,
                 const float* __restrict__ k_all,
                 const float* __restrict__ v_all,
                 const int*   __restrict__ adj,
                 float* __restrict__ ctx) {
    const int e    = blockIdx.x;
    const int h    = threadIdx.x >> 5;   // head
    const int lane = threadIdx.x & 31;

    __shared__ int   s_adj[K_NBR];
    __shared__ float s_q[D_DIM];
    __shared__ float s_attn[H_HEADS][K_NBR];

    if (threadIdx.x < K_NBR) s_adj[threadIdx.x] = adj[(size_t)e * K_NBR + threadIdx.x];
    s_q[threadIdx.x] = q[(size_t)e * D_DIM + threadIdx.x];
    __syncthreads();

    // --- scores: lane = neighbor ---
    const int nb = s_adj[lane];
    const float4* k4 = (const float4*)(k_all + (size_t)nb * D_DIM + h * DH_DIM);
    const float4* q4 = (const float4*)(s_q + h * DH_DIM);
    float s = 0.f;
    #pragma unroll
    for (int d = 0; d < DH_DIM / 4; ++d) {
        float4 kk = k4[d];
        float4 qq = q4[d];
        s += qq.x * kk.x + qq.y * kk.y + qq.z * kk.z + qq.w * kk.w;
    }
    s *= 0.17677669529663687f;           // 1/sqrt(32)

    // --- softmax across the 32 lanes of this wave ---
    float m = s;
    #pragma unroll
    for (int off = 16; off > 0; off >>= 1) m = fmaxf(m, __shfl_xor(m, off, 32));
    float ex = __expf(s - m);
    float sum = ex;
    #pragma unroll
    for (int off = 16; off > 0; off >>= 1) sum += __shfl_xor(sum, off, 32);
    s_attn[h][lane] = ex / sum;
    __syncthreads();

    // --- ctx: lane = head-dim, loop neighbors (coalesced 128B per step) ---
    float acc = 0.f;
    #pragma unroll 8
    for (int j = 0; j < K_NBR; ++j)
        acc += s_attn[h][j] * v_all[(size_t)s_adj[j] * D_DIM + h * DH_DIM + lane];
    ctx[(size_t)e * D_DIM + h * DH_DIM + lane] = acc;
}

// ---------------------------------------------------------------------------
// Kernel 3: fused  out = (gelu(ctx @ Wo @ W1 + b1)) @ W2 + b2
// grid.x = E/64, block = 256 (8 waves), WMMA tile chain through LDS.
// ---------------------------------------------------------------------------
__device__ __forceinline__ float gelu_tanh(float x) {
    float x3 = x * x * x;
    float t = tanhf(0.7978845608028654f * (x + 0.044715f * x3));
    return 0.5f * x * (1.f + t);
}

__global__ void __launch_bounds__(256)
mlp_kernel(const float* __restrict__ ctx,
           const float* __restrict__ Wo,
           const float* __restrict__ W1,
           const float* __restrict__ b1,
           const float* __restrict__ W2,
           const float* __restrict__ b2,
           float* __restrict__ out) {
    __shared__ float sA[ROWS_PER_BLK * D_DIM];
    __shared__ float sB[ROWS_PER_BLK * D_DIM];
    const int row0 = blockIdx.x * ROWS_PER_BLK;

    if (threadIdx.x < 32) {
        tdm_load_tile_f32(ctx + (size_t)row0 * D_DIM, sA, ROWS_PER_BLK);
        __builtin_amdgcn_s_wait_tensorcnt(0);
    }
    __syncthreads();

    const int wave = threadIdx.x >> 5;
    const int lane = threadIdx.x & 31;
    const int n0   = wave * 16;
    const int g    = lane >> 4;
    const int ncol = n0 + (lane & 15);

    // edge_out = ctx @ Wo  -> sB
    {
        v8f acc[MT_TILES] = {};
        tile_gemm_4m(sA, Wo, n0, lane, acc);
        #pragma unroll
        for (int mt = 0; mt < MT_TILES; ++mt)
            #pragma unroll
            for (int i = 0; i < 8; ++i)
                sB[(mt * 16 + i + 8 * g) * D_DIM + ncol] = acc[mt][i];
    }
    __syncthreads();

    // h = gelu(edge_out @ W1 + b1) -> sA (all sA reads completed before barrier)
    {
        v8f acc[MT_TILES] = {};
        tile_gemm_4m(sB, W1, n0, lane, acc);
        float bias = b1[ncol];
        #pragma unroll
        for (int mt = 0; mt < MT_TILES; ++mt)
            #pragma unroll
            for (int i = 0; i < 8; ++i)
                sA[(mt * 16 + i + 8 * g) * D_DIM + ncol] = gelu_tanh(acc[mt][i] + bias);
    }
    __syncthreads();

    // logits = h @ W2 + b2   (128x10; waves 0..3 each compute one 16-row tile)
    if (wave < MT_TILES) {
        const int mt   = wave;
        const int mrow = lane & 15;
        const int n    = lane & 15;
        v8f acc = {};
        #pragma unroll 8
        for (int k0 = 0; k0 < D_DIM; k0 += 4) {
            v2f a, b;
            a[0] = sA[(mt * 16 + mrow) * D_DIM + k0 + 2 * g];
            a[1] = sA[(mt * 16 + mrow) * D_DIM + k0 + 2 * g + 1];
            b[0] = (n < C_CLS) ? W2[(size_t)(k0 + 2 * g)     * C_CLS + n] : 0.f;
            b[1] = (n < C_CLS) ? W2[(size_t)(k0 + 2 * g + 1) * C_CLS + n] : 0.f;
            acc = wmma_f32_step(a, b, acc);
        }
        if (n < C_CLS) {
            float bias = b2[n];
            #pragma unroll
            for (int i = 0; i < 8; ++i)
                out[(size_t)(row0 + mt * 16 + i + 8 * g) * C_CLS + n] = acc[i] + bias;
        }
    }
}

// ---------------------------------------------------------------------------
// Launch
// ---------------------------------------------------------------------------
extern "C" void kernel_launch(void* const* d_in, const int* in_sizes, int n_in,
                              void* d_out, int out_size, void* d_ws, size_t ws_size,
                              hipStream_t stream) {
    const float* X  = (const float*)d_in[0];
    const float* Wq = (const float*)d_in[1];
    const float* Wk = (const float*)d_in[2];
    const float* Wv = (const float*)d_in[3];
    const float* Wo = (const float*)d_in[4];
    const float* W1 = (const float*)d_in[5];
    const float* b1 = (const float*)d_in[6];
    const float* W2 = (const float*)d_in[7];
    const float* b2 = (const float*)d_in[8];
    const int*  adj = (const int*)d_in[9];
    float* out = (float*)d_out;

    const size_t ED = (size_t)E_EDGES * D_DIM;
    float* ws = (float*)d_ws;
    float* q     = ws;
    float* k_all = ws + ED;
    float* v_all = ws + 2 * ED;
    float* ctxb  = ws + 3 * ED;

    qkv_proj_kernel<<<E_EDGES / ROWS_PER_BLK, 256, 0, stream>>>(X, Wq, Wk, Wv, q, k_all, v_all);
    edge_attn_kernel<<<E_EDGES, 128, 0, stream>>>(q, k_all, v_all, adj, ctxb);
    mlp_kernel<<<E_EDGES / ROWS_PER_BLK, 256, 0, stream>>>(ctxb, Wo, W1, b1, W2, b2, out);
}